// STICKYKVCache_LayerWise_25082609009241
// MI455X (gfx1250) — compile-verified
//
#include <hip/hip_runtime.h>
#include <hip/hip_bf16.h>
#include <stdint.h>

// Problem constants (fixed by the reference: H=32, S=2048, SINK=4, OMEGA=32)
#define H_DIM   32
#define S_DIM   2048
#define SINKC   4
#define MAXW    10000
#define NWIN    63     // (2048-4)/32
#define NPROMPT 62     // (2048-32-4)/32
#define QCHUNK  32     // query chunks per head
#define QPB     64     // queries per block (2048/32)
#define PSTRIDE 128    // padded per-(h,chunk) partial stride (126 floats used)

typedef __attribute__((ext_vector_type(2))) float v2f;
typedef __attribute__((ext_vector_type(8))) float v8f;

// ---------------- kernel 1: constant background of [32, 10000, 6] ----------
__global__ void thc_fill(float* __restrict__ out) {
  int idx = blockIdx.x * blockDim.x + threadIdx.x;
  if (idx >= H_DIM * MAXW * 6) return;
  int c = idx % 6;
  int w = (idx / 6) % MAXW;
  float v;
  if (c < 2)       v = 0.0f;                                     // local_history default
  else if (c == 4) v = (w < NWIN) ? (float)w : __builtin_nanf(""); // window index
  else             v = __builtin_nanf("");                       // window_scores default
  out[idx] = v;
}

// ---------------- kernel 2: HBM-streaming pass via async-to-LDS ------------
// Block = one (head, 64-query chunk). Thread t owns 8 contiguous columns
// starting at 4+8t; a lane-quad owns one 32-wide window (w = t/4, w<63).
// Data is staged global->LDS with the CDNA5 async copy engine, 4-row ring.
__global__ void __launch_bounds__(256) thc_pass1(const float* __restrict__ attn,
                                                 float* __restrict__ partial) {
  __shared__ __align__(16) float smem[4][256 * 8];   // 4-deep ring, 8 KB/row
  const int t  = threadIdx.x;
  const int qc = blockIdx.x;          // 0..31 query chunk
  const int h  = blockIdx.y;          // 0..31 head
  // t in [0,252): columns [4+8t, 12+8t). Lanes 252..255 load a safe in-bounds
  // dummy slice (their quad would be window 63, which is discarded).
  const int col = (t < 252) ? (SINKC + 8 * t) : 2016;
  const float thr = 32.0f / 2048.0f;  // OMEGA / seq_len (exact)

  const float* base = attn + ((size_t)h * S_DIM + (size_t)qc * QPB) * S_DIM + col;
  // LDS byte offset of this thread's slot (generic-addr low 32 bits == LDS addr)
  const uint32_t lds_base = (uint32_t)(uintptr_t)&smem[0][t * 8];

  auto issue = [&](int r) {
    uint64_t g = (uint64_t)(base + (size_t)r * S_DIM);
    uint32_t l = lds_base + (uint32_t)(r & 3) * (uint32_t)(256 * 8 * sizeof(float));
    // inst_offset applies to both LDS and global sides -> two B128s move 32 B
    asm volatile("global_load_async_to_lds_b128 %0, %1, off th:TH_LOAD_NT"
                 :: "v"(l), "v"(g) : "memory");
    asm volatile("global_load_async_to_lds_b128 %0, %1, off offset:16 th:TH_LOAD_NT"
                 :: "v"(l), "v"(g) : "memory");
  };

  float mag = 0.0f, hits = 0.0f;
  auto consume = [&](int i) {
    const float* sp = &smem[i & 3][t * 8];
    float4 x = *(const float4*)sp;
    float4 y = *(const float4*)(sp + 4);
    float v = ((x.x + x.y) + (x.z + x.w)) + ((y.x + y.y) + (y.z + y.w));
    v += __shfl_xor(v, 1, 32);        // quad reduce (wave32)
    v += __shfl_xor(v, 2, 32);
    mag  += v;                        // window magnitude accumulation
    hits += (v > thr) ? 1.0f : 0.0f;  // strict > per reference
  };

  issue(0); issue(1); issue(2);                       // prime 3-row lookahead
  for (int i = 0; i < QPB - 3; ++i) {
    issue(i + 3);                                     // 8 async ops in flight
    asm volatile("s_wait_asynccnt 6" ::: "memory");   // oldest row landed
    consume(i);
  }
  asm volatile("s_wait_asynccnt 4" ::: "memory"); consume(QPB - 3);
  asm volatile("s_wait_asynccnt 2" ::: "memory"); consume(QPB - 2);
  asm volatile("s_wait_asynccnt 0" ::: "memory"); consume(QPB - 1);

  if (t < 252 && (t & 3) == 0) {      // quad leader, window w = t/4 < 63
    int w = t >> 2;
    float* p = partial + ((size_t)h * QCHUNK + qc) * PSTRIDE + 2 * w;
    p[0] = mag;
    p[1] = hits;
  }
}

// ---------------- kernel 3: WMMA cross-chunk reduction + scatter -----------
// Per head: S[j] = sum over 32 chunks of partial[h][qc][j], j = 2*w + metric.
// A = ones(16x4) makes D[m][n] = sum_k B[k][n]; 8 accumulating
// v_wmma_f32_16x16x4_f32 reduce all 32 chunks for 16 columns at a time.
// (A=ones => result independent of the exact K<->lane mapping of B.)
__global__ void thc_reduce_wmma(const float* __restrict__ partial,
                                float* __restrict__ out) {
  const int h    = blockIdx.x;        // 32 blocks
  const int lane = threadIdx.x;       // one full wave32, EXEC all ones
  const int n    = lane & 15;
  const int kb   = (lane < 16) ? 0 : 2;
  const float* Ph = partial + (size_t)h * QCHUNK * PSTRIDE;

  v2f a; a.x = 1.0f; a.y = 1.0f;      // ones A-matrix (16x4)

  for (int jt = 0; jt < 8; ++jt) {    // 8 column tiles cover j = 0..127
    const int jb = jt * 16;
    v8f c = {};
#pragma unroll
    for (int g = 0; g < 8; ++g) {     // 8 K-steps x 4 chunks = 32 chunks
      const int qc = g * 4 + kb;
      v2f b;
      b.x = Ph[(size_t)qc       * PSTRIDE + jb + n];
      b.y = Ph[(size_t)(qc + 1) * PSTRIDE + jb + n];
      c = __builtin_amdgcn_wmma_f32_16x16x4_f32(false, a, false, b,
                                                (short)0, c, false, false);
    }
    float s = c[0];                   // lanes 0..15 hold D[0][n] = S[jb+n]
    int j = jb + n;
    if (lane < 16 && j < 2 * NWIN) {  // j >= 126 is padding -> discard
      int w = j >> 1;
      float* o = out + ((size_t)h * MAXW + w) * 6;
      if ((j & 1) == 0) { o[2] = s; if (w < NPROMPT) o[0] = s; }  // mag
      else              { o[3] = s; if (w < NPROMPT) o[1] = s; }  // hits
    }
  }
}

// ---------------- launch ---------------------------------------------------
extern "C" void kernel_launch(void* const* d_in, const int* in_sizes, int n_in,
                              void* d_out, int out_size, void* d_ws, size_t ws_size,
                              hipStream_t stream) {
  (void)in_sizes; (void)n_in; (void)out_size; (void)ws_size;
  const float* attn = (const float*)d_in[0];   // [1,32,2048,2048] f32
  float* out = (float*)d_out;                  // [32,10000,6] f32
  float* partial = (float*)d_ws;               // 32*32*128*4 = 512 KiB scratch

  const int total = H_DIM * MAXW * 6;
  thc_fill<<<(total + 255) / 256, 256, 0, stream>>>(out);
  thc_pass1<<<dim3(QCHUNK, H_DIM), 256, 0, stream>>>(attn, partial);
  thc_reduce_wmma<<<H_DIM, 32, 0, stream>>>(partial, out);
}